// augGNN_71116068487689
// MI455X (gfx1250) — compile-verified
//
#include <hip/hip_runtime.h>
#include <math.h>

#define NN 50000
#define NE 600000
#define BN_EPS 1e-5f

typedef __attribute__((ext_vector_type(2))) float v2f;
typedef __attribute__((ext_vector_type(8))) float v8f;

// D = A(16x4,f32) * B(4x16,f32) + C(16x16,f32), wave32.
// A layout: lane m=lane&15, K = (lane>>4)*2 + vgpr. B layout: lane n=lane&15, same K pairing.
// C/D: col = lane&15, row = vgpr + (lane>>4)*8.
__device__ __forceinline__ v8f wmma_k4(v2f a, v2f b, v8f c) {
  return __builtin_amdgcn_wmma_f32_16x16x4_f32(false, a, false, b, (short)0, c,
                                               false, false);
}

// ---------------- BN folding: per-channel scale/bias ----------------
__global__ void k_precompute(const float* w1a, const float* b1a,
                             const float* g1, const float* be1,
                             const float* m1, const float* v1,
                             const float* b2a, const float* g2,
                             const float* be2, const float* m2,
                             const float* v2p, float* cst) {
  int c = threadIdx.x;  // 256 threads
  if (c < 256) {
    float inv = g1[c] * rsqrtf(v1[c] + BN_EPS);
    cst[c]       = inv * w1a[c];                     // c1scale
    cst[256 + c] = inv * (b1a[c] - m1[c]) + be1[c];  // c1bias
  }
  if (c < 64) {
    float inv = g2[c] * rsqrtf(v2p[c] + BN_EPS);
    cst[512 + c] = inv;                              // c2scale
    cst[576 + c] = inv * (b2a[c] - m2[c]) + be2[c];  // c2bias
  }
}

// ---------------- scalar aggregation: s = x[:,col] + segsum ----------------
__global__ void k_init_s(const float* x, int col, float* s) {
  int i = blockIdx.x * blockDim.x + threadIdx.x;
  if (i < NN) s[i] = x[2 * i + col];
}

__global__ void k_scatter_s(const float* x, const int* ei, int col, float* s) {
  int e = blockIdx.x * blockDim.x + threadIdx.x;
  if (e >= NE) return;
  int src = ei[e], dst = ei[NE + e];
  atomicAdd(&s[dst], x[2 * src + col]);
}

// ---------------- GEMM1: h2[N,128] = relu(bn1(s*w1a+b1a)) @ w1b + b1b ------
__global__ void k_gemm_mlp1(const float* __restrict__ s,
                            const float* __restrict__ c1s,
                            const float* __restrict__ c1b,
                            const float* __restrict__ w1b,
                            const float* __restrict__ b1b,
                            float* __restrict__ h2) {
  int lane = threadIdx.x & 31, wave = threadIdx.x >> 5;
  int mr = lane & 15, half = lane >> 4;
  int row0 = blockIdx.x * 16;
  int colL = (blockIdx.y * 4 + wave) * 16 + mr;
  float sv = s[row0 + mr];
  v8f acc = {};
#pragma unroll 4
  for (int k0 = 0; k0 < 256; k0 += 4) {
    int ka = k0 + half * 2;
    v2f a, b;
    a.x = fmaxf(sv * c1s[ka] + c1b[ka], 0.f);
    a.y = fmaxf(sv * c1s[ka + 1] + c1b[ka + 1], 0.f);
    b.x = w1b[ka * 128 + colL];
    b.y = w1b[(ka + 1) * 128 + colL];
    acc = wmma_k4(a, b, acc);
  }
  float bb = b1b[colL];
#pragma unroll
  for (int m = 0; m < 8; ++m)
    h2[(row0 + m + half * 8) * 128 + colL] = acc[m] + bb;
}

// ---------------- edge scatter on h2 (128 channels) ----------------
__global__ void k_scatter_h2(const float* __restrict__ h2, const int* ei,
                             float* __restrict__ h2s) {
  int t = blockIdx.x * blockDim.x + threadIdx.x;
  int e = t >> 5;
  if (e >= NE) return;
  int c4 = (t & 31) * 4;
  int src = ei[e], dst = ei[NE + e];
  const float4 v = *reinterpret_cast<const float4*>(&h2[src * 128 + c4]);
  atomicAdd(&h2s[dst * 128 + c4 + 0], v.x);
  atomicAdd(&h2s[dst * 128 + c4 + 1], v.y);
  atomicAdd(&h2s[dst * 128 + c4 + 2], v.z);
  atomicAdd(&h2s[dst * 128 + c4 + 3], v.w);
}

// ---------------- GEMM2: h3[N,64] = relu(bn2(h2s @ w2a + b2a)) -------------
__global__ void k_gemm_bn2(const float* __restrict__ h2s,
                           const float* __restrict__ w2a,
                           const float* __restrict__ c2s,
                           const float* __restrict__ c2b,
                           float* __restrict__ h3) {
  int lane = threadIdx.x & 31, wave = threadIdx.x >> 5;
  int mr = lane & 15, half = lane >> 4;
  int row0 = blockIdx.x * 16;
  int colL = wave * 16 + mr;
  v8f acc = {};
#pragma unroll 4
  for (int k0 = 0; k0 < 128; k0 += 4) {
    int ka = k0 + half * 2;
    v2f a, b;
    a.x = h2s[(row0 + mr) * 128 + ka];
    a.y = h2s[(row0 + mr) * 128 + ka + 1];
    b.x = w2a[ka * 64 + colL];
    b.y = w2a[(ka + 1) * 64 + colL];
    acc = wmma_k4(a, b, acc);
  }
  float sc = c2s[colL], bb = c2b[colL];
#pragma unroll
  for (int m = 0; m < 8; ++m)
    h3[(row0 + m + half * 8) * 64 + colL] = fmaxf(acc[m] * sc + bb, 0.f);
}

// ---------------- GEMM3: e[N,64] = h3 @ w2b + b2b ----------------
__global__ void k_gemm_e(const float* __restrict__ h3,
                         const float* __restrict__ w2b,
                         const float* __restrict__ b2b,
                         float* __restrict__ eo) {
  int lane = threadIdx.x & 31, wave = threadIdx.x >> 5;
  int mr = lane & 15, half = lane >> 4;
  int row0 = blockIdx.x * 16;
  int colL = wave * 16 + mr;
  v8f acc = {};
#pragma unroll 4
  for (int k0 = 0; k0 < 64; k0 += 4) {
    int ka = k0 + half * 2;
    v2f a, b;
    a.x = h3[(row0 + mr) * 64 + ka];
    a.y = h3[(row0 + mr) * 64 + ka + 1];
    b.x = w2b[ka * 64 + colL];
    b.y = w2b[(ka + 1) * 64 + colL];
    acc = wmma_k4(a, b, acc);
  }
  float bb = b2b[colL];
#pragma unroll
  for (int m = 0; m < 8; ++m)
    eo[(row0 + m + half * 8) * 64 + colL] = acc[m] + bb;
}

// ---------------- GEMM4: ff[N,64] = [e1|e2] @ V ----------------
__global__ void k_gemm_ff(const float* __restrict__ e1,
                          const float* __restrict__ e2,
                          const float* __restrict__ V,
                          float* __restrict__ ff) {
  int lane = threadIdx.x & 31, wave = threadIdx.x >> 5;
  int mr = lane & 15, half = lane >> 4;
  int row0 = blockIdx.x * 16;
  int colL = wave * 16 + mr;
  int r = row0 + mr;
  v8f acc = {};
#pragma unroll 4
  for (int k0 = 0; k0 < 128; k0 += 4) {
    int ka = k0 + half * 2;
    v2f a, b;
    a.x = (ka < 64) ? e1[r * 64 + ka] : e2[r * 64 + ka - 64];
    int kb = ka + 1;
    a.y = (kb < 64) ? e1[r * 64 + kb] : e2[r * 64 + kb - 64];
    b.x = V[ka * 64 + colL];
    b.y = V[kb * 64 + colL];
    acc = wmma_k4(a, b, acc);
  }
#pragma unroll
  for (int m = 0; m < 8; ++m)
    ff[(row0 + m + half * 8) * 64 + colL] = acc[m];
}

// ---------------- W transpose: Wt[c*64+k] = W[k*4096+c] ----------------
__global__ void k_transpose_W(const float* __restrict__ W, float* __restrict__ Wt) {
  int t = blockIdx.x * blockDim.x + threadIdx.x;  // 262144 threads
  Wt[t] = W[(t & 63) * 4096 + (t >> 6)];          // coalesced writes
}

// ------- GEMM5: Bkn[k][i] = sum_c (e1[i][c>>6]*e2[i][c&63]) * Wt[c][k] -----
// 80 rows per block (50000 = 625*80): 5 accumulators per wave so every Wt
// load feeds 5 WMMAs; e1 factor hoisted out of the inner-b loop (c = a*64+b).
__global__ void k_gemm_bilinear(const float* __restrict__ e1,
                                const float* __restrict__ e2,
                                const float* __restrict__ Wt,
                                float* __restrict__ Bkn) {
  __shared__ float s1[64 * 80];  // transposed: s1[a*80+node]
  __shared__ float s2[64 * 80];
  int row0 = blockIdx.x * 80;
  for (int idx = threadIdx.x; idx < 80 * 16; idx += 128) {
    int node = idx >> 4, a4 = (idx & 15) * 4;
    float4 u = *reinterpret_cast<const float4*>(&e1[(row0 + node) * 64 + a4]);
    s1[(a4 + 0) * 80 + node] = u.x;
    s1[(a4 + 1) * 80 + node] = u.y;
    s1[(a4 + 2) * 80 + node] = u.z;
    s1[(a4 + 3) * 80 + node] = u.w;
    float4 w = *reinterpret_cast<const float4*>(&e2[(row0 + node) * 64 + a4]);
    s2[(a4 + 0) * 80 + node] = w.x;
    s2[(a4 + 1) * 80 + node] = w.y;
    s2[(a4 + 2) * 80 + node] = w.z;
    s2[(a4 + 3) * 80 + node] = w.w;
  }
  __syncthreads();
  int lane = threadIdx.x & 31, wave = threadIdx.x >> 5;
  int mr = lane & 15, half = lane >> 4;
  int colL = wave * 16 + mr;  // k index, 4 waves cover k=0..63
  v8f acc[5] = {{}, {}, {}, {}, {}};
  for (int aa = 0; aa < 64; ++aa) {
    float e1v[5];
#pragma unroll
    for (int t = 0; t < 5; ++t) e1v[t] = s1[aa * 80 + mr + 16 * t];
    const float* wrow = Wt + aa * 4096;  // c = aa*64 + b
#pragma unroll 4
    for (int b0 = 0; b0 < 64; b0 += 4) {
      int bb = b0 + half * 2;
      v2f b;
      b.x = wrow[bb * 64 + colL];
      b.y = wrow[(bb + 1) * 64 + colL];
#pragma unroll
      for (int t = 0; t < 5; ++t) {
        v2f a;
        a.x = e1v[t] * s2[bb * 80 + mr + 16 * t];
        a.y = e1v[t] * s2[(bb + 1) * 80 + mr + 16 * t];
        acc[t] = wmma_k4(a, b, acc[t]);
      }
    }
  }
#pragma unroll
  for (int t = 0; t < 5; ++t)
#pragma unroll
    for (int m = 0; m < 8; ++m)
      Bkn[colL * NN + row0 + t * 16 + m + half * 8] = acc[t][m];  // [k][n] layout
}

// ---------------- final: scramble + tanh + head + log_softmax --------------
__global__ void k_final(const float* __restrict__ Bkn, const float* __restrict__ ff,
                        const float* __restrict__ bvec, const float* __restrict__ w4,
                        const float* __restrict__ b4, const float* __restrict__ wl2,
                        const float* __restrict__ bl2, float* __restrict__ out) {
  int i = blockIdx.x * blockDim.x + threadIdx.x;
  if (i >= NN) return;
  float sumb = 0.f;
#pragma unroll
  for (int c = 0; c < 64; ++c) sumb += bvec[c];
  float z[6];
#pragma unroll
  for (int c = 0; c < 6; ++c) z[c] = b4[c];
  for (int j = 0; j < 64; ++j) {
    int f = i * 64 + j;                      // torch-faithful (k,N)->(N,k) reshape
    int kk = f / NN, nn = f - kk * NN;
    float hj = tanhf(Bkn[kk * NN + nn] + sumb + ff[i * 64 + j]);
#pragma unroll
    for (int c = 0; c < 6; ++c) z[c] += hj * w4[j * 6 + c];
  }
  float y[6];
#pragma unroll
  for (int c = 0; c < 6; ++c) {
    float acc = bl2[c];
#pragma unroll
    for (int j = 0; j < 6; ++j) acc += fmaxf(z[j], 0.f) * wl2[j * 6 + c];
    y[c] = acc;
  }
  float mx = y[0];
#pragma unroll
  for (int c = 1; c < 6; ++c) mx = fmaxf(mx, y[c]);
  float se = 0.f;
#pragma unroll
  for (int c = 0; c < 6; ++c) se += expf(y[c] - mx);
  float ls = mx + logf(se);
#pragma unroll
  for (int c = 0; c < 6; ++c) out[i * 6 + c] = y[c] - ls;
}

extern "C" void kernel_launch(void* const* d_in, const int* in_sizes, int n_in,
                              void* d_out, int out_size, void* d_ws, size_t ws_size,
                              hipStream_t stream) {
  const float* x    = (const float*)d_in[0];
  const float* w1a  = (const float*)d_in[1];
  const float* b1a  = (const float*)d_in[2];
  const float* g1   = (const float*)d_in[3];
  const float* be1  = (const float*)d_in[4];
  const float* m1   = (const float*)d_in[5];
  const float* v1   = (const float*)d_in[6];
  const float* w1b  = (const float*)d_in[7];
  const float* b1b  = (const float*)d_in[8];
  const float* w2a  = (const float*)d_in[9];
  const float* b2a  = (const float*)d_in[10];
  const float* g2   = (const float*)d_in[11];
  const float* be2  = (const float*)d_in[12];
  const float* m2   = (const float*)d_in[13];
  const float* v2p  = (const float*)d_in[14];
  const float* w2b  = (const float*)d_in[15];
  const float* b2b  = (const float*)d_in[16];
  const float* W    = (const float*)d_in[17];
  const float* V    = (const float*)d_in[18];
  const float* bvec = (const float*)d_in[19];
  const float* w4   = (const float*)d_in[20];
  const float* b4   = (const float*)d_in[21];
  const float* wl2  = (const float*)d_in[22];
  const float* bl2  = (const float*)d_in[23];
  const int*   ei   = (const int*)d_in[24];
  (void)in_sizes; (void)n_in; (void)out_size; (void)ws_size;

  float* ws  = (float*)d_ws;                 // ~90 MB workspace layout (floats)
  float* s   = ws;                           // 50000  (padded to 50048)
  float* h2  = ws + 50048;                   // 6,400,000
  float* h2s = h2 + 6400000;                 // 6,400,000
  float* h3  = h2s + 6400000;                // 3,200,000
  float* e1  = h3 + 3200000;                 // 3,200,000
  float* e2  = e1 + 3200000;                 // 3,200,000
  float* cst = e2 + 3200000;                 // 640
  float* ff  = h2;                           // alias (h2 dead after h3)
  float* Bkn = h2s;                          // alias
  float* Wt  = h3;                           // alias (1 MB within h3)

  k_precompute<<<1, 256, 0, stream>>>(w1a, b1a, g1, be1, m1, v1, b2a, g2, be2,
                                      m2, v2p, cst);

  for (int blk = 0; blk < 2; ++blk) {
    int col = (blk == 0) ? 1 : 0;            // e1 from x[:,1], e2 from x[:,0]
    float* eo = (blk == 0) ? e1 : e2;
    k_init_s<<<(NN + 255) / 256, 256, 0, stream>>>(x, col, s);
    k_scatter_s<<<(NE + 255) / 256, 256, 0, stream>>>(x, ei, col, s);
    k_gemm_mlp1<<<dim3(NN / 16, 2), 128, 0, stream>>>(s, cst, cst + 256, w1b,
                                                      b1b, h2);
    hipMemcpyAsync(h2s, h2, (size_t)NN * 128 * sizeof(float),
                   hipMemcpyDeviceToDevice, stream);
    k_scatter_h2<<<(NE * 32) / 256, 256, 0, stream>>>(h2, ei, h2s);
    k_gemm_bn2<<<dim3(NN / 16, 1), 128, 0, stream>>>(h2s, w2a, cst + 512,
                                                     cst + 576, h3);
    k_gemm_e<<<dim3(NN / 16, 1), 128, 0, stream>>>(h3, w2b, b2b, eo);
  }

  k_transpose_W<<<262144 / 256, 256, 0, stream>>>(W, Wt);
  k_gemm_ff<<<dim3(NN / 16, 1), 128, 0, stream>>>(e1, e2, V, ff);
  k_gemm_bilinear<<<NN / 80, 128, 0, stream>>>(e1, e2, Wt, Bkn);
  k_final<<<(NN + 255) / 256, 256, 0, stream>>>(Bkn, ff, bvec, w4, b4, wl2, bl2,
                                                (float*)d_out);
}